// LISTA_17471926960563
// MI455X (gfx1250) — compile-verified
//
#include <hip/hip_runtime.h>
#include <hip/hip_bf16.h>
#include <math.h>

typedef float v2f __attribute__((ext_vector_type(2)));
typedef float v8f __attribute__((ext_vector_type(8)));

#define NUM_ATOMS 512
#define NUM_DIMS  128
#define LDZ 516   // padded LDS row stride for 512-wide tiles (even, ==4 mod 64)
#define LDX 132   // padded LDS row stride for 128-wide tiles (even, ==4 mod 64)
#define TM  64    // rows (pixels) per workgroup

// D = A(16x4,f32) * B(4x16,f32) + C(16x16,f32)
static __device__ __forceinline__ v8f wmma_f32(v2f a, v2f b, v8f c) {
    return __builtin_amdgcn_wmma_f32_16x16x4_f32(false, a, false, b, (short)0, c,
                                                 false, false);
}

static __device__ __forceinline__ v8f vzero8() {
    v8f z = {0.f, 0.f, 0.f, 0.f, 0.f, 0.f, 0.f, 0.f};
    return z;
}

// ---------------------------------------------------------------------------
// Kernel 1: S = I - (Dict^T Dict)/L   (512x512, trivially small: 67 MFLOP)
// ---------------------------------------------------------------------------
__global__ __launch_bounds__(256) void lista_gram_s(const float* __restrict__ Dict,
                                                    const float* __restrict__ Lp,
                                                    float* __restrict__ S) {
    int idx = blockIdx.x * 256 + threadIdx.x;   // 512*512 threads
    int i = idx >> 9;
    int j = idx & 511;
    float s = 0.f;
    #pragma unroll 8
    for (int d = 0; d < NUM_DIMS; ++d)
        s = fmaf(Dict[d * NUM_ATOMS + i], Dict[d * NUM_ATOMS + j], s);
    float invL = 1.0f / Lp[0];
    S[idx] = (i == j ? 1.0f : 0.0f) - s * invL;
}

// ---------------------------------------------------------------------------
// Kernel 2: per-WG persistent LISTA tile.
//   WG = 256 threads (8 wave32s), handles one (b, h-pair): 64 pixel-rows.
//   LDS: z[64][LDZ], y/L[64][LDZ], xt/recon[64][LDX], lthr[64]  (~298 KB)
//   Iteration GEMM per wave: 4 row-blocks x 4 N-tiles -> 16 acc tiles,
//   4 ds A-loads + 8 global B-loads per 16 WMMA.
// ---------------------------------------------------------------------------
__global__ __launch_bounds__(256) void lista_main(const float* __restrict__ x,
                                                  const float* __restrict__ Dict,
                                                  const float* __restrict__ alpha,
                                                  const float* __restrict__ Lp,
                                                  const int* __restrict__ nip,
                                                  const float* __restrict__ S,
                                                  float* __restrict__ out) {
    extern __shared__ float smem[];
    float* zs   = smem;                  // TM*LDZ
    float* yls  = zs + TM * LDZ;         // TM*LDZ  (y * 1/L)
    float* xts  = yls + TM * LDZ;        // TM*LDX  (x tile, later recon tile)
    float* lthr = xts + TM * LDX;        // TM      (alpha/L per pixel row)

    const int wg   = blockIdx.x;         // 0..511  ->  m0 = wg*64
    const int b    = wg >> 4;            // batch
    const int h2   = wg & 15;            // h pair: h = 2*h2 + hh
    const int pixb = h2 * 64;            // (h*32+w) base for this WG
    const int tid  = threadIdx.x;
    const int lane = tid & 31;           // wave32
    const int warp = tid >> 5;           // 0..7

    const float invL = 1.0f / Lp[0];
    const int   NI   = nip[0];

    // ---- load x[b,:,2h2:2h2+2,:] -> xts[row][d] (coalesced 32-float runs) ----
    const float* xb = x + (size_t)b * (NUM_DIMS * 1024) + pixb;
    for (int d = warp; d < NUM_DIMS; d += 8) {
        xts[lane * LDX + d]        = xb[(size_t)d * 1024 + lane];
        xts[(32 + lane) * LDX + d] = xb[(size_t)d * 1024 + 32 + lane];
    }
    if (tid < TM) lthr[tid] = alpha[pixb + tid] * invL;
    __syncthreads();

    const int mrow  = lane & 15;          // A-frag row within 16-row block
    const int khalf = (lane >> 4) << 1;   // 0 or 2 : A/B K-pair select
    const int ncol  = lane & 15;          // B/C-frag column within tile
    const int crow  = (lane >> 4) << 3;   // C-frag +8 row offset for hi lanes

    // =======================================================================
    // y = xt @ Dict  (K=128); z = soft(y, alpha/L); yls = y/L
    // wave: 4 row-blocks x 4 N-tiles (N-range = warp*64 .. +64)
    // =======================================================================
    {
        v8f acc[4][4];
        #pragma unroll
        for (int rb = 0; rb < 4; ++rb)
            #pragma unroll
            for (int nt = 0; nt < 4; ++nt) acc[rb][nt] = vzero8();

        for (int k = 0; k < NUM_DIMS; k += 4) {
            v2f a[4];
            #pragma unroll
            for (int rb = 0; rb < 4; ++rb) {
                const float* p = &xts[(rb * 16 + mrow) * LDX + k + khalf];
                a[rb].x = p[0]; a[rb].y = p[1];
            }
            const float* Bp = &Dict[(k + khalf) * NUM_ATOMS + warp * 64 + ncol];
            __builtin_prefetch(Bp + 8 * NUM_ATOMS);   // global_prefetch_b8
            #pragma unroll
            for (int nt = 0; nt < 4; ++nt) {
                v2f bf;
                bf.x = Bp[nt * 16];
                bf.y = Bp[nt * 16 + NUM_ATOMS];
                #pragma unroll
                for (int rb = 0; rb < 4; ++rb)
                    acc[rb][nt] = wmma_f32(a[rb], bf, acc[rb][nt]);
            }
        }
        #pragma unroll
        for (int rb = 0; rb < 4; ++rb)
            #pragma unroll
            for (int nt = 0; nt < 4; ++nt)
                #pragma unroll
                for (int v = 0; v < 8; ++v) {
                    int row = rb * 16 + v + crow;
                    int col = warp * 64 + nt * 16 + ncol;
                    float yv = acc[rb][nt][v];
                    yls[row * LDZ + col] = yv * invL;
                    float t = lthr[row];
                    zs[row * LDZ + col] =
                        copysignf(fmaxf(fabsf(yv) - t, 0.0f), yv);
                }
    }
    __syncthreads();

    // =======================================================================
    // 16 x: z = soft(z @ S + y/L)   (K=512, S is L2-resident)
    // =======================================================================
    for (int it = 0; it < NI; ++it) {
        v8f acc[4][4];
        #pragma unroll
        for (int rb = 0; rb < 4; ++rb)
            #pragma unroll
            for (int nt = 0; nt < 4; ++nt) acc[rb][nt] = vzero8();

        for (int k = 0; k < NUM_ATOMS; k += 4) {
            v2f a[4];
            #pragma unroll
            for (int rb = 0; rb < 4; ++rb) {
                const float* p = &zs[(rb * 16 + mrow) * LDZ + k + khalf];
                a[rb].x = p[0]; a[rb].y = p[1];
            }
            const float* Bp = &S[(k + khalf) * NUM_ATOMS + warp * 64 + ncol];
            __builtin_prefetch(Bp + 8 * NUM_ATOMS);   // next K-panel into WGP$
            #pragma unroll
            for (int nt = 0; nt < 4; ++nt) {
                v2f bf;
                bf.x = Bp[nt * 16];
                bf.y = Bp[nt * 16 + NUM_ATOMS];
                #pragma unroll
                for (int rb = 0; rb < 4; ++rb)
                    acc[rb][nt] = wmma_f32(a[rb], bf, acc[rb][nt]);
            }
        }
        __syncthreads();   // all reads of old z done
        #pragma unroll
        for (int rb = 0; rb < 4; ++rb)
            #pragma unroll
            for (int nt = 0; nt < 4; ++nt)
                #pragma unroll
                for (int v = 0; v < 8; ++v) {
                    int row = rb * 16 + v + crow;
                    int col = warp * 64 + nt * 16 + ncol;
                    float val = acc[rb][nt][v] + yls[row * LDZ + col];
                    float t = lthr[row];
                    zs[row * LDZ + col] =
                        copysignf(fmaxf(fabsf(val) - t, 0.0f), val);
                }
        __syncthreads();
    }

    // =======================================================================
    // x_recon = z @ Dict^T  (K=512, N=128); stage in xts for coalesced store
    // wave: 1 row-block (warp>>1) x 4 N-tiles (64-col half = warp&1)
    // =======================================================================
    {
        const int rb = warp >> 1;
        const int n0 = (warp & 1) * 64;
        v8f acc[4];
        #pragma unroll
        for (int nt = 0; nt < 4; ++nt) acc[nt] = vzero8();

        for (int k = 0; k < NUM_ATOMS; k += 4) {
            v2f a;
            {
                const float* p = &zs[(rb * 16 + mrow) * LDZ + k + khalf];
                a.x = p[0]; a.y = p[1];
            }
            #pragma unroll
            for (int nt = 0; nt < 4; ++nt) {
                const float* dp =
                    &Dict[(size_t)(n0 + nt * 16 + ncol) * NUM_ATOMS + k + khalf];
                v2f bf;
                bf.x = dp[0];    // Dict^T[k][n] = Dict[n*512 + k] (contiguous k)
                bf.y = dp[1];
                acc[nt] = wmma_f32(a, bf, acc[nt]);
            }
        }
        #pragma unroll
        for (int nt = 0; nt < 4; ++nt)
            #pragma unroll
            for (int v = 0; v < 8; ++v) {
                int row = rb * 16 + v + crow;
                int col = n0 + nt * 16 + ncol;
                xts[row * LDX + col] = acc[nt][v];
            }
    }
    __syncthreads();

    // ---- out[0] : z  -> [B, A, H, W] ----
    float* outz = out + (size_t)b * (NUM_ATOMS * 1024) + pixb;
    for (int a = warp; a < NUM_ATOMS; a += 8) {
        outz[(size_t)a * 1024 + lane]      = zs[lane * LDZ + a];
        outz[(size_t)a * 1024 + 32 + lane] = zs[(32 + lane) * LDZ + a];
    }

    // ---- out[1] : x_recon -> [B, D, H, W] ----
    float* outr = out + (size_t)16777216 + (size_t)b * (NUM_DIMS * 1024) + pixb;
    for (int d = warp; d < NUM_DIMS; d += 8) {
        outr[(size_t)d * 1024 + lane]      = xts[lane * LDX + d];
        outr[(size_t)d * 1024 + 32 + lane] = xts[(32 + lane) * LDX + d];
    }
}

// ---------------------------------------------------------------------------
// Kernel 3: pass-through Dict copy (3rd tuple output)
// ---------------------------------------------------------------------------
__global__ __launch_bounds__(256) void lista_copy_dict(const float* __restrict__ Dict,
                                                       float* __restrict__ out) {
    int i = blockIdx.x * 256 + threadIdx.x;
    out[i] = Dict[i];
}

extern "C" void kernel_launch(void* const* d_in, const int* in_sizes, int n_in,
                              void* d_out, int out_size, void* d_ws, size_t ws_size,
                              hipStream_t stream) {
    const float* x     = (const float*)d_in[0];   // [32,128,32,32]
    const float* Dict  = (const float*)d_in[1];   // [128,512]
    const float* alpha = (const float*)d_in[2];   // [1,32,32,1]
    const float* Lp    = (const float*)d_in[3];   // scalar
    const int*   nip   = (const int*)d_in[4];     // scalar (16)
    float* out = (float*)d_out;
    float* S   = (float*)d_ws;                    // 512*512 floats = 1 MB

    // S = I - (D^T D)/L
    lista_gram_s<<<(512 * 512) / 256, 256, 0, stream>>>(Dict, Lp, S);

    // main LISTA: 512 WGs (one per (b, h-pair)), 256 threads, ~298 KB LDS
    size_t shmem = (size_t)(2 * TM * LDZ + TM * LDX + TM) * sizeof(float);
    lista_main<<<512, 256, shmem, stream>>>(x, Dict, alpha, Lp, nip, S, out);

    // Dict pass-through at tail of d_out (offset 16777216 + 4194304)
    lista_copy_dict<<<65536 / 256, 256, 0, stream>>>(Dict, out + 20971520);
}